// VectorQuantize_59313498357808
// MI455X (gfx1250) — compile-verified
//
#include <hip/hip_runtime.h>

typedef __attribute__((ext_vector_type(2))) float v2f;
typedef __attribute__((ext_vector_type(8))) float v8f;

#define N_TOK 32768
#define M_CB  4096
#define D_DIM 256
#define NTILE (M_CB / 16)     // 256 codebook column tiles
#define BPAD  260             // LDS row stride in dwords (padded: 4*lr banks)

// ---------------------------------------------------------------------------
// Kernel 1: e2[m] = ||embed[m]||^2   (one wave per codebook row)
// ---------------------------------------------------------------------------
__global__ void vq_e2_kernel(const float* __restrict__ embed,
                             float* __restrict__ e2) {
  int wave = (blockIdx.x * blockDim.x + threadIdx.x) >> 5;
  int lane = threadIdx.x & 31;
  if (wave >= M_CB) return;
  const float* row = embed + (size_t)wave * D_DIM;
  float s = 0.f;
#pragma unroll
  for (int k = 0; k < D_DIM / 32; ++k) {
    float v = row[lane + 32 * k];
    s += v * v;
  }
#pragma unroll
  for (int m = 16; m >= 1; m >>= 1) s += __shfl_xor(s, m, 32);
  if (lane == 0) e2[wave] = s;
}

// ---------------------------------------------------------------------------
// Kernel 2: fused distance-GEMM + row argmin, V_WMMA_F32_16X16X4_F32.
//
// Block = 128 threads = 4 waves, __launch_bounds__(128,1).
// Each wave owns a 16-row x tile (A resident in VGPRs, 128 VGPRs).  All 4
// waves sweep the same 16-column codebook tiles, double-buffered in LDS and
// fed by GLOBAL_LOAD_ASYNC_TO_LDS_B128 (zero VGPR cost, ASYNCcnt-tracked)
// while the WMMA stream computes the current tile -> no register prefetch,
// no scratch spills (round-2 failure mode).
//
// f32 16x16x4 layouts (ISA 7.12.2):
//   A (16x4, MxK):  lane {half,lr}: M=lr, VGPR0/1 = K = 2*half + {0,1}
//   B (4x16, KxN):  lane {half,lr}: N=lr, VGPR0/1 = K = 2*half + {0,1}
//   C/D (16x16):    lane {half,lr}: col N=lr, VGPR v: row M = v + 8*half
// ---------------------------------------------------------------------------
__global__ __launch_bounds__(128, 1)
void vq_argmin_kernel(const float* __restrict__ x,
                      const float* __restrict__ embed,
                      const float* __restrict__ e2,
                      float* __restrict__ idx_out) {
  __shared__ float lds_b[2][16 * BPAD];       // 2 x 16.6 KB staged tiles

  const int lane = threadIdx.x & 31;
  const int wib  = threadIdx.x >> 5;          // wave in block (0..3)
  const int half = lane >> 4;                 // 0 or 1
  const int lr   = lane & 15;
  const int r0   = (blockIdx.x * 4 + wib) * 16;

  // Cooperative-copy mapping: thread t covers row (t>>3), cols (t&7)*32..+31
  // = one contiguous 128B segment per thread (8 x b128 beats).
  const int ld_row = threadIdx.x >> 3;
  const int ld_seg = threadIdx.x & 7;

  // Per-thread base addresses for the async tile copy.  The instruction
  // IOFFSET is added to BOTH the global and LDS addresses (ISA 15.18), and
  // both streams advance 16B per beat, so offset:i*16 covers the segment.
  const float* gsrc0 = embed + (size_t)ld_row * D_DIM + ld_seg * 32;
  unsigned lds_dst[2];
  lds_dst[0] = (unsigned)(size_t)&lds_b[0][ld_row * BPAD + ld_seg * 32];
  lds_dst[1] = (unsigned)(size_t)&lds_b[1][ld_row * BPAD + ld_seg * 32];

  // ---- Resident A tile: 16 rows x 256 K of x -----------------------------
  v2f a[64];
  const float* xrow = x + (size_t)(r0 + lr) * D_DIM + 2 * half;
#pragma unroll
  for (int kc = 0; kc < 64; ++kc)
    a[kc] = *(const v2f*)(xrow + kc * 4);

  // ---- Async-stage tile 0 into LDS buffer 0 ------------------------------
#pragma unroll
  for (int i = 0; i < 8; ++i)
    asm volatile("global_load_async_to_lds_b128 %0, %1, off offset:%2"
                 :: "v"(lds_dst[0]), "v"(gsrc0), "i"(i * 16) : "memory");
  asm volatile("s_wait_asynccnt 0x0" ::: "memory");
  __syncthreads();

  float minv[8];
  int   mini[8];
#pragma unroll
  for (int v = 0; v < 8; ++v) { minv[v] = 3.4e38f; mini[v] = 0; }

  for (int t = 0; t < NTILE; ++t) {
    const int buf = t & 1;

    // -- issue async copy of tile t+1 into the other buffer ----------------
    if (t + 1 < NTILE) {
      const float* gsrc = gsrc0 + (size_t)(t + 1) * 16 * D_DIM;
      const unsigned dst = lds_dst[buf ^ 1];
#pragma unroll
      for (int i = 0; i < 8; ++i)
        asm volatile("global_load_async_to_lds_b128 %0, %1, off offset:%2"
                     :: "v"(dst), "v"(gsrc), "i"(i * 16) : "memory");
    }

    // -- 16x16x256 tile MAC: 64 WMMAs, two independent accumulator chains --
    const float* bp = &lds_b[buf][lr * BPAD + 2 * half];
    v8f c0 = {}, c1 = {};
#pragma unroll
    for (int kc = 0; kc < 64; kc += 2) {
      v2f b0 = *(const v2f*)(bp + kc * 4);
      v2f b1 = *(const v2f*)(bp + kc * 4 + 4);
      c0 = __builtin_amdgcn_wmma_f32_16x16x4_f32(
               false, a[kc],     false, b0, (short)0, c0, false, false);
      c1 = __builtin_amdgcn_wmma_f32_16x16x4_f32(
               false, a[kc + 1], false, b1, (short)0, c1, false, false);
    }

    // -- epilogue: d = e2[col] - 2*dot ; running (min, argmin) per row -----
    const float e2v = e2[t * 16 + lr];
    const int   col = t * 16 + lr;
#pragma unroll
    for (int v = 0; v < 8; ++v) {
      float d = __builtin_fmaf(-2.0f, c0[v] + c1[v], e2v);
      if (d < minv[v]) { minv[v] = d; mini[v] = col; }
    }

    // -- pipeline sync: my asyncs landed; all waves done with both buffers -
    asm volatile("s_wait_asynccnt 0x0" ::: "memory");
    __syncthreads();
  }

  // ---- min+index reduction across the 16 lanes sharing each row ----------
#pragma unroll
  for (int v = 0; v < 8; ++v) {
#pragma unroll
    for (int m = 8; m >= 1; m >>= 1) {
      float ov = __shfl_xor(minv[v], m, 32);
      int   oi = __shfl_xor(mini[v], m, 32);
      if (ov < minv[v] || (ov == minv[v] && oi < mini[v])) {
        minv[v] = ov; mini[v] = oi;
      }
    }
    if (lr == 0) {
      int row = r0 + v + 8 * half;
      idx_out[row] = (float)mini[v];       // exact for idx < 2^24
    }
  }
}

// ---------------------------------------------------------------------------
// Kernel 3: gather q = embed[idx], write quantized (= q forward), accumulate
// sum((q-x)^2) and cluster counts.  One 256-thread block per token row.
// ---------------------------------------------------------------------------
__global__ void vq_gather_loss_kernel(const float* __restrict__ x,
                                      const float* __restrict__ embed,
                                      const float* __restrict__ idx_f,
                                      float* __restrict__ quant,
                                      float* __restrict__ counts,
                                      float* __restrict__ sqerr) {
  __shared__ float red[8];
  const int row  = blockIdx.x;
  const int t    = threadIdx.x;                 // 256 == D_DIM
  const int code = (int)idx_f[row];

  float xv = x[(size_t)row * D_DIM + t];
  float qv = embed[(size_t)code * D_DIM + t];
  quant[(size_t)row * D_DIM + t] = qv;

  float d = qv - xv;
  d *= d;
#pragma unroll
  for (int m = 16; m >= 1; m >>= 1) d += __shfl_xor(d, m, 32);
  if ((t & 31) == 0) red[t >> 5] = d;
  __syncthreads();
  if (t == 0) {
    float s = 0.f;
#pragma unroll
    for (int i = 0; i < 8; ++i) s += red[i];
    atomicAdd(sqerr, s);
    atomicAdd(&counts[code], 1.0f);
  }
}

// ---------------------------------------------------------------------------
// Kernel 4: loss + perplexity finalize (single block).
//   loss = (1 + 0.25) * sqerr / (N*D)
//   perplexity = exp(-sum(p * log(p + 1e-10))),  p = counts / N
// ---------------------------------------------------------------------------
__global__ void vq_finalize_kernel(const float* __restrict__ counts,
                                   const float* __restrict__ sqerr,
                                   float* __restrict__ out_tail) {
  __shared__ float red[256];
  const int t = threadIdx.x;
  float s = 0.f;
  for (int i = t; i < M_CB; i += 256) {
    float p = counts[i] * (1.0f / (float)N_TOK);
    s += p * logf(p + 1e-10f);
  }
  red[t] = s;
  __syncthreads();
  for (int off = 128; off >= 1; off >>= 1) {
    if (t < off) red[t] += red[t + off];
    __syncthreads();
  }
  if (t == 0) {
    out_tail[0] = 1.25f * sqerr[0] / (float)(N_TOK * D_DIM);  // loss
    out_tail[1] = expf(-red[0]);                              // perplexity
  }
}

// ---------------------------------------------------------------------------
extern "C" void kernel_launch(void* const* d_in, const int* in_sizes, int n_in,
                              void* d_out, int out_size, void* d_ws, size_t ws_size,
                              hipStream_t stream) {
  const float* x     = (const float*)d_in[0];   // [N, D]
  const float* embed = (const float*)d_in[1];   // [M, D]

  float* out   = (float*)d_out;
  float* quant = out;                                   // [N*D]
  float* idx_f = out + (size_t)N_TOK * D_DIM;           // [N]
  float* tail  = idx_f + N_TOK;                         // loss, perplexity

  float* e2     = (float*)d_ws;                         // [M]
  float* counts = e2 + M_CB;                            // [M]
  float* sqerr  = counts + M_CB;                        // [1]

  // zero counts + sqerr each call (graph-capturable memset node)
  hipMemsetAsync(counts, 0, (M_CB + 1) * sizeof(float), stream);

  vq_e2_kernel<<<M_CB / 4, 128, 0, stream>>>(embed, e2);
  vq_argmin_kernel<<<N_TOK / 64, 128, 0, stream>>>(x, embed, e2, idx_f);
  vq_gather_loss_kernel<<<N_TOK, 256, 0, stream>>>(x, embed, idx_f, quant,
                                                   counts, sqerr);
  vq_finalize_kernel<<<1, 256, 0, stream>>>(counts, sqerr, tail);
}